// EpisodicMemory_19662360281122
// MI455X (gfx1250) — compile-verified
//
#include <hip/hip_runtime.h>
#include <hip/hip_bf16.h>
#include <math.h>

typedef __attribute__((ext_vector_type(2))) float v2f;
typedef __attribute__((ext_vector_type(4))) float v4f;
typedef __attribute__((ext_vector_type(8))) float v8f;

#define B_ 64
#define S_ 2048
#define D_ 512
#define TOPK_ 8
#define NCHUNKS_ 16
#define RPC_ (S_ / NCHUNKS_)   // rows per chunk = 128
#define ATT_SCALE 0.044194173824159216f  // 1/sqrt(512)

// ---------------------------------------------------------------------------
// Kernel 1: C = A @ W^T (+bias), M=64 fixed, via V_WMMA_F32_16X16X4_F32.
// One wave (32 threads) computes one 16x16 tile of C.
// A: M x K row-major; W: N x K row-major (so B-matrix = W^T, element (k,n)=W[n][k]).
// Fragment layout per ISA 7.12.2 (32-bit A 16x4): lanes 0-15 = rows, lane-half
// selects K pair; symmetric for B (lanes = N cols). C/D: VGPR r holds
// row (r + 8*(lane>=16)), col = lane%16.
// ---------------------------------------------------------------------------
__global__ __launch_bounds__(32) void wmma_gemm_at_kernel(
    const float* __restrict__ A, const float* __restrict__ W,
    const float* __restrict__ bias, float* __restrict__ C,
    int N, int K)
{
    const int mt   = blockIdx.x;
    const int nt   = blockIdx.y;
    const int lane = threadIdx.x;
    const int l16  = lane & 15;
    const int hi   = lane >> 4;      // 0 or 1: selects K pair

    const float* arow = A + (size_t)(mt * 16 + l16) * K + 2 * hi;
    const float* brow = W + (size_t)(nt * 16 + l16) * K + 2 * hi;

    v8f acc = {};
    for (int k = 0; k < K; k += 4) {
        v2f a = *(const v2f*)(arow + k);
        v2f b = *(const v2f*)(brow + k);
        acc = __builtin_amdgcn_wmma_f32_16x16x4_f32(
            /*neg_a=*/false, a, /*neg_b=*/false, b,
            /*c_mod=*/(short)0, acc, /*reuse_a=*/false, /*reuse_b=*/false);
    }

    const int col = nt * 16 + l16;
    const float bv = bias ? bias[col] : 0.0f;
#pragma unroll
    for (int r = 0; r < 8; ++r) {
        const int row = mt * 16 + r + 8 * hi;
        C[(size_t)row * N + col] = acc[r] + bv;
    }
}

// ---------------------------------------------------------------------------
// Kernel 2: per-batch top-8 of w_logits + softmax over the 8 values.
// Tie-break: smaller index wins (matches jax.lax.top_k first-occurrence).
// Also emits write_strength = gate.
// ---------------------------------------------------------------------------
__global__ __launch_bounds__(256) void topk_kernel(
    const float* __restrict__ wl, const float* __restrict__ gate,
    int* __restrict__ oidx, float* __restrict__ ow,
    float* __restrict__ wstr)
{
    __shared__ float vals[S_];
    __shared__ float rv[256];
    __shared__ int   ri[256];
    __shared__ float topv[TOPK_];
    __shared__ int   topi[TOPK_];

    const int b = blockIdx.x, t = threadIdx.x;
    for (int i = t; i < S_; i += 256) vals[i] = wl[(size_t)b * S_ + i];
    __syncthreads();

    for (int sel = 0; sel < TOPK_; ++sel) {
        float best = -INFINITY; int bi = 0x7fffffff;
        for (int i = t; i < S_; i += 256) {
            const float v = vals[i];
            if (v > best) { best = v; bi = i; }
        }
        rv[t] = best; ri[t] = bi;
        __syncthreads();
        for (int off = 128; off > 0; off >>= 1) {
            if (t < off) {
                const float vo = rv[t + off]; const int io = ri[t + off];
                if (vo > rv[t] || (vo == rv[t] && io < ri[t])) { rv[t] = vo; ri[t] = io; }
            }
            __syncthreads();
        }
        if (t == 0) { topv[sel] = rv[0]; topi[sel] = ri[0]; vals[ri[0]] = -INFINITY; }
        __syncthreads();
    }

    if (t == 0) {
        const float m = topv[0];             // descending -> max first
        float e[TOPK_], sum = 0.0f;
#pragma unroll
        for (int j = 0; j < TOPK_; ++j) { e[j] = __expf(topv[j] - m); sum += e[j]; }
        const float inv = 1.0f / sum;
#pragma unroll
        for (int j = 0; j < TOPK_; ++j) {
            ow[b * TOPK_ + j]   = e[j] * inv;
            oidx[b * TOPK_ + j] = topi[j];
        }
        wstr[b] = gate[b];
    }
}

// ---------------------------------------------------------------------------
// Kernel 3: stream mem_k once (NT loads/stores: one-touch, bigger than L2).
// Per s-row (one wave): att[b,s] = q.k / sqrt(D), and
// mem_k2 = k + g*w_s*(write_k - k)  (w_s sparse: nonzero on 8 rows only).
// ---------------------------------------------------------------------------
__global__ __launch_bounds__(256) void stream_k_kernel(
    const float* __restrict__ mem_k, const float* __restrict__ q,
    const float* __restrict__ wk, const float* __restrict__ gate,
    const int* __restrict__ idx, const float* __restrict__ wsp,
    float* __restrict__ att, float* __restrict__ memk2)
{
    const int b     = blockIdx.y;
    const int chunk = blockIdx.x;
    const int wave  = threadIdx.x >> 5;
    const int lane  = threadIdx.x & 31;

    const float g = gate[b];
    int   tki[TOPK_]; float tkw[TOPK_];
#pragma unroll
    for (int j = 0; j < TOPK_; ++j) { tki[j] = idx[b * TOPK_ + j]; tkw[j] = wsp[b * TOPK_ + j]; }

    v4f qv[4], kv[4];
#pragma unroll
    for (int j = 0; j < 4; ++j) {
        qv[j] = *(const v4f*)(q  + (size_t)b * D_ + j * 128 + lane * 4);
        kv[j] = *(const v4f*)(wk + (size_t)b * D_ + j * 128 + lane * 4);
    }

    const int s0 = chunk * RPC_;
    for (int s = s0 + wave; s < s0 + RPC_; s += 8) {
        const float* krow = mem_k + ((size_t)b * S_ + s) * D_;
        float*       orow = memk2 + ((size_t)b * S_ + s) * D_;
        __builtin_prefetch(krow + 8 * D_ + lane * 16, 0, 0);   // next row for this wave

        float ws = 0.0f;
#pragma unroll
        for (int j = 0; j < TOPK_; ++j) if (tki[j] == s) ws = tkw[j];
        const float gw = g * ws;

        float dot = 0.0f;
#pragma unroll
        for (int j = 0; j < 4; ++j) {
            const v4f kvv =
                __builtin_nontemporal_load((const v4f*)(krow + j * 128 + lane * 4));
            dot += kvv.x * qv[j].x + kvv.y * qv[j].y + kvv.z * qv[j].z + kvv.w * qv[j].w;
            v4f o;
            o.x = kvv.x + gw * (kv[j].x - kvv.x);
            o.y = kvv.y + gw * (kv[j].y - kvv.y);
            o.z = kvv.z + gw * (kv[j].z - kvv.z);
            o.w = kvv.w + gw * (kv[j].w - kvv.w);
            __builtin_nontemporal_store(o, (v4f*)(orow + j * 128 + lane * 4));
        }
#pragma unroll
        for (int off = 16; off > 0; off >>= 1) dot += __shfl_xor(dot, off, 32);
        if (lane == 0) att[(size_t)b * S_ + s] = dot * ATT_SCALE;
    }
}

// ---------------------------------------------------------------------------
// Kernel 4: w_read = softmax(att) over S per batch row.
// ---------------------------------------------------------------------------
__global__ __launch_bounds__(256) void softmax_kernel(
    const float* __restrict__ att, float* __restrict__ wread)
{
    __shared__ float red[256];
    const int b = blockIdx.x, t = threadIdx.x;
    const float* row = att + (size_t)b * S_;

    float m = -INFINITY;
    for (int i = t; i < S_; i += 256) m = fmaxf(m, row[i]);
    red[t] = m; __syncthreads();
    for (int off = 128; off > 0; off >>= 1) { if (t < off) red[t] = fmaxf(red[t], red[t + off]); __syncthreads(); }
    m = red[0]; __syncthreads();

    float sum = 0.0f;
    for (int i = t; i < S_; i += 256) sum += __expf(row[i] - m);
    red[t] = sum; __syncthreads();
    for (int off = 128; off > 0; off >>= 1) { if (t < off) red[t] += red[t + off]; __syncthreads(); }
    const float inv = 1.0f / red[0];

    for (int i = t; i < S_; i += 256) wread[(size_t)b * S_ + i] = __expf(row[i] - m) * inv;
}

// ---------------------------------------------------------------------------
// Kernel 5: stream mem_v once (NT loads/stores). Writes mem_v2 (sparse gated
// update) and accumulates read_out partial = sum_s w_read[s]*v[s,:] per chunk.
// ---------------------------------------------------------------------------
__global__ __launch_bounds__(256) void stream_v_kernel(
    const float* __restrict__ mem_v, const float* __restrict__ wread,
    const float* __restrict__ wv, const float* __restrict__ gate,
    const int* __restrict__ idx, const float* __restrict__ wsp,
    float* __restrict__ memv2, float* __restrict__ partial)
{
    __shared__ float sacc[8 * D_];   // 16 KB: per-wave partial vectors

    const int b     = blockIdx.y;
    const int chunk = blockIdx.x;
    const int wave  = threadIdx.x >> 5;
    const int lane  = threadIdx.x & 31;

    const float g = gate[b];
    int   tki[TOPK_]; float tkw[TOPK_];
#pragma unroll
    for (int j = 0; j < TOPK_; ++j) { tki[j] = idx[b * TOPK_ + j]; tkw[j] = wsp[b * TOPK_ + j]; }

    v4f vv[4];
#pragma unroll
    for (int j = 0; j < 4; ++j)
        vv[j] = *(const v4f*)(wv + (size_t)b * D_ + j * 128 + lane * 4);

    v4f acc[4] = {};
    const int s0 = chunk * RPC_;
    for (int s = s0 + wave; s < s0 + RPC_; s += 8) {
        const float* vrow = mem_v + ((size_t)b * S_ + s) * D_;
        float*       orow = memv2 + ((size_t)b * S_ + s) * D_;
        __builtin_prefetch(vrow + 8 * D_ + lane * 16, 0, 0);

        const float wr = wread[(size_t)b * S_ + s];
        float ws = 0.0f;
#pragma unroll
        for (int j = 0; j < TOPK_; ++j) if (tki[j] == s) ws = tkw[j];
        const float gw = g * ws;

#pragma unroll
        for (int j = 0; j < 4; ++j) {
            const v4f vcur =
                __builtin_nontemporal_load((const v4f*)(vrow + j * 128 + lane * 4));
            acc[j].x += wr * vcur.x; acc[j].y += wr * vcur.y;
            acc[j].z += wr * vcur.z; acc[j].w += wr * vcur.w;
            v4f o;
            o.x = vcur.x + gw * (vv[j].x - vcur.x);
            o.y = vcur.y + gw * (vv[j].y - vcur.y);
            o.z = vcur.z + gw * (vv[j].z - vcur.z);
            o.w = vcur.w + gw * (vv[j].w - vcur.w);
            __builtin_nontemporal_store(o, (v4f*)(orow + j * 128 + lane * 4));
        }
    }

#pragma unroll
    for (int j = 0; j < 4; ++j)
        *(v4f*)(&sacc[wave * D_ + j * 128 + lane * 4]) = acc[j];
    __syncthreads();

    for (int col = threadIdx.x; col < D_; col += 256) {
        float sum = 0.0f;
#pragma unroll
        for (int w = 0; w < 8; ++w) sum += sacc[w * D_ + col];
        partial[((size_t)b * NCHUNKS_ + chunk) * D_ + col] = sum;
    }
}

// ---------------------------------------------------------------------------
// Kernel 6: read_out[b,col] = sum over chunks of partials.
// ---------------------------------------------------------------------------
__global__ __launch_bounds__(256) void reduce_kernel(
    const float* __restrict__ partial, float* __restrict__ read_out)
{
    const int i = blockIdx.x * 256 + threadIdx.x;   // [0, B*D)
    const int b = i >> 9, col = i & (D_ - 1);
    float sum = 0.0f;
#pragma unroll
    for (int c = 0; c < NCHUNKS_; ++c)
        sum += partial[((size_t)b * NCHUNKS_ + c) * D_ + col];
    read_out[i] = sum;
}

// ---------------------------------------------------------------------------
extern "C" void kernel_launch(void* const* d_in, const int* in_sizes, int n_in,
                              void* d_out, int out_size, void* d_ws, size_t ws_size,
                              hipStream_t stream) {
    const float* s     = (const float*)d_in[0];
    const float* wvec  = (const float*)d_in[1];
    const float* mem_k = (const float*)d_in[2];
    const float* mem_v = (const float*)d_in[3];
    const float* gate  = (const float*)d_in[4];
    const float* Wq    = (const float*)d_in[5];
    const float* Wl    = (const float*)d_in[6];
    const float* bl    = (const float*)d_in[7];
    const float* Wk    = (const float*)d_in[8];
    const float* Wv    = (const float*)d_in[9];
    // d_in[10] = topk (fixed 8)

    // workspace layout (floats)
    float* ws      = (float*)d_ws;
    float* q       = ws;                       // B*D
    float* wk      = q    + B_ * D_;           // B*D
    float* wvv     = wk   + B_ * D_;           // B*D
    float* wl      = wvv  + B_ * D_;           // B*S
    float* att     = wl   + B_ * S_;           // B*S
    float* wread   = att  + B_ * S_;           // B*S
    float* tkw     = wread + B_ * S_;          // B*8
    int*   tki     = (int*)(tkw + B_ * TOPK_); // B*8
    float* partial = (float*)(tki + B_ * TOPK_); // B*NCHUNKS*D

    // output layout: read_out | mem_k2 | mem_v2 | write_strength
    float* out_read = (float*)d_out;
    float* out_k2   = out_read + (size_t)B_ * D_;
    float* out_v2   = out_k2 + (size_t)B_ * S_ * D_;
    float* out_wstr = out_v2 + (size_t)B_ * S_ * D_;

    const dim3 w32(32);
    wmma_gemm_at_kernel<<<dim3(B_ / 16, D_ / 16), w32, 0, stream>>>(s,    Wq, nullptr, q,   D_, D_);
    wmma_gemm_at_kernel<<<dim3(B_ / 16, S_ / 16), w32, 0, stream>>>(s,    Wl, bl,      wl,  S_, D_);
    wmma_gemm_at_kernel<<<dim3(B_ / 16, D_ / 16), w32, 0, stream>>>(wvec, Wk, nullptr, wk,  D_, D_);
    wmma_gemm_at_kernel<<<dim3(B_ / 16, D_ / 16), w32, 0, stream>>>(wvec, Wv, nullptr, wvv, D_, D_);

    topk_kernel<<<B_, 256, 0, stream>>>(wl, gate, tki, tkw, out_wstr);

    stream_k_kernel<<<dim3(NCHUNKS_, B_), 256, 0, stream>>>(mem_k, q, wk, gate, tki, tkw, att, out_k2);

    softmax_kernel<<<B_, 256, 0, stream>>>(att, wread);

    stream_v_kernel<<<dim3(NCHUNKS_, B_), 256, 0, stream>>>(mem_v, wread, wvv, gate, tki, tkw, out_v2, partial);

    reduce_kernel<<<(B_ * D_) / 256, 256, 0, stream>>>(partial, out_read);
}